// diagmulsel_50646254354648
// MI455X (gfx1250) — compile-verified
//
#include <hip/hip_runtime.h>
#include <hip/hip_bf16.h>
#include <cstdint>

// out[r, c] = scale[r] * A[r, c],  scale[r] = r < n_sel ? 2*sigmoid(d[r]) : 1
//
// Pure streaming kernel: 256 MB in + 256 MB out @ 23.3 TB/s => ~23 us floor.
// Data path: global -> LDS via CDNA5 async DMA loads (ASYNCcnt), scale in
// VALU, non-temporal b128 stores out (no reuse; don't pollute L2).

#define COLS    8192
#define TPB     256
#define STAGES  (COLS / (TPB * 4))     // 8 stages of 1024 floats
#define CHUNK_B (TPB * 16)             // 4096 bytes per stage

typedef float v4f __attribute__((ext_vector_type(4)));

__global__ __launch_bounds__(TPB) void diagmulsel_rowscale_kernel(
    const float* __restrict__ A,
    const float* __restrict__ d,
    float* __restrict__ out,
    int n_sel) {
  // Whole row staged in LDS: 8 stages * 256 lanes * 16B = 32 KB.
  __shared__ v4f tile[STAGES * TPB];

  const int row = blockIdx.x;
  const int tid = threadIdx.x;

  // Uniform per-row scale (scalar load of d[row], negligible traffic).
  float s = 1.0f;
  if (row < n_sel) {
    const float dv = d[row];
    s = 2.0f / (1.0f + __expf(-dv));   // 2*sigmoid(dv)
  }

  const float* rowA = A + (size_t)row * (size_t)COLS;
  v4f* rowO = (v4f*)(out + (size_t)row * (size_t)COLS);

  // Per-lane LDS byte address, derived from the REAL pointer so `tile` is
  // captured by the asm (otherwise LLVM proves tile is never written, folds
  // the readback loads to NaN, and deletes the ds_load/mul chain — observed
  // in round 1). Flat shared-aperture addresses carry the LDS byte offset in
  // their low 32 bits ({SHARED_BASE,16'b0} aperture base is in the high 32).
  const uint32_t loff = (uint32_t)(uintptr_t)(&tile[tid]);
  const uint32_t goff = (uint32_t)tid * 16u;  // global byte offset vs rowA

  // Issue all async global->LDS copies for this row (ASYNCcnt += STAGES).
  // The 24-bit IOFFSET immediate is added to BOTH the LDS destination and
  // the global source address, so one VGPR pair serves all 8 stages.
#pragma unroll
  for (int st = 0; st < STAGES; ++st) {
    asm volatile(
        "global_load_async_to_lds_b128 %0, %1, %2 offset:%3"
        :
        : "v"(loff), "v"(goff), "s"(rowA), "n"(st * CHUNK_B)
        : "memory");
  }

  // Wait for this wave's async copies to land in LDS (in-order completion;
  // each lane reads back only the 16B it deposited -> no barrier needed).
#if __has_builtin(__builtin_amdgcn_s_wait_asynccnt)
  __builtin_amdgcn_s_wait_asynccnt(0);
#else
  asm volatile("s_wait_asynccnt 0" ::: "memory");
#endif

  // Scale and stream out with non-temporal 128-bit stores.
#pragma unroll
  for (int st = 0; st < STAGES; ++st) {
    v4f v = tile[st * TPB + tid];   // ds_load_b128
    v = v * s;
    __builtin_nontemporal_store(v, &rowO[st * TPB + tid]);  // b128, th:NT
  }
}

extern "C" void kernel_launch(void* const* d_in, const int* in_sizes, int n_in,
                              void* d_out, int out_size, void* d_ws, size_t ws_size,
                              hipStream_t stream) {
  const float* A = (const float*)d_in[0];
  const float* d = (const float*)d_in[1];
  float* out = (float*)d_out;

  const int n_sel = in_sizes[1];               // 2048
  const int rows  = in_sizes[0] / COLS;        // 8192

  diagmulsel_rowscale_kernel<<<dim3(rows), dim3(TPB), 0, stream>>>(A, d, out, n_sel);
}